// Normalize_8985071583848
// MI455X (gfx1250) — compile-verified
//
#include <hip/hip_runtime.h>
#include <stdint.h>

#define N_EDGES 1600000
#define N_NODES 50000
#define D 32

// Per-wave LDS staging capacity in edge rows (multiple of 4). Segment lengths
// ~ Poisson(32); P(len > 96) ~ 1e-19 per node, and a global-memory tail path
// handles any length anyway.
#define CAP 96
#define WAVES_PER_BLOCK 4   // 4 waves * 96 rows * 128B = 48 KB static LDS

typedef __attribute__((ext_vector_type(4))) float f32x4;

static __device__ __forceinline__ f32x4 make4(float v) {
    f32x4 r = {v, v, v, v};
    return r;
}
static __device__ __forceinline__ f32x4 vmax4(f32x4 a, f32x4 b) {
    f32x4 r;
    r.x = fmaxf(a.x, b.x); r.y = fmaxf(a.y, b.y);
    r.z = fmaxf(a.z, b.z); r.w = fmaxf(a.w, b.w);
    return r;
}
static __device__ __forceinline__ f32x4 vexp4(f32x4 a) {
    f32x4 r;
    r.x = __expf(a.x); r.y = __expf(a.y);
    r.z = __expf(a.z); r.w = __expf(a.w);
    return r;
}
static __device__ __forceinline__ f32x4 shflxor4(f32x4 v, int mask) {
    f32x4 r;
    r.x = __shfl_xor(v.x, mask, 32);
    r.y = __shfl_xor(v.y, mask, 32);
    r.z = __shfl_xor(v.z, mask, 32);
    r.w = __shfl_xor(v.w, mask, 32);
    return r;
}

// ---------------------------------------------------------------------------
// Kernel 1: segment offsets from the sorted index column.
// seg_start[n] = first edge e with idx[e] >= n; seg_start[N_NODES] = N_EDGES.
// ---------------------------------------------------------------------------
__global__ void k_seg_offsets(const int* __restrict__ sel, int* __restrict__ seg_start) {
    int e = blockIdx.x * blockDim.x + threadIdx.x;
    if (e >= N_EDGES) return;
    int cur  = sel[(size_t)e * 6 + 4];
    int prev = (e == 0) ? -1 : sel[(size_t)(e - 1) * 6 + 4];
    for (int n = prev + 1; n <= cur; ++n) seg_start[n] = e;
    if (e == N_EDGES - 1) {
        for (int n = cur + 1; n <= N_NODES; ++n) seg_start[n] = N_EDGES;
    }
}

// ---------------------------------------------------------------------------
// Kernel 2: one wave32 per node. phase = lane>>3 picks one of 4 edge rows per
// group, (lane&7)*4 picks the 16B dim chunk => each group is 4 rows x b128 =
// 512B coalesced per wave. Segment rows are DMA-staged into LDS once with
// gfx1250 async-to-LDS loads (ASYNCcnt); the 3 softmax passes then read LDS.
// ---------------------------------------------------------------------------
__global__ __launch_bounds__(WAVES_PER_BLOCK * 32)
void k_seg_softmax(const float* __restrict__ x,
                   const int* __restrict__ seg_start,
                   float* __restrict__ out) {
    __shared__ __align__(16) float smem[WAVES_PER_BLOCK * CAP * D];

    const int lane = threadIdx.x & 31;
    const int wave = threadIdx.x >> 5;
    const int node = blockIdx.x * WAVES_PER_BLOCK + wave;
    if (node >= N_NODES) return;            // uniform per wave

    const int start = seg_start[node];
    const int end   = seg_start[node + 1];
    const int len   = end - start;
    if (len <= 0) return;                   // uniform per wave

    const int phase = lane >> 3;            // 0..3: row within group of 4
    const int dofs  = (lane & 7) * 4;       // float offset of this lane's 16B chunk
    const int cached = (len < CAP) ? len : CAP;
    const int full   = cached & ~3;         // full groups of 4 rows in LDS

    float* wls = &smem[wave * CAP * D];
    const float* lp = wls + phase * D + dofs;                 // this lane's LDS cursor
    const float* gp = x + ((size_t)(start + phase)) * D + dofs; // this lane's global cursor
    // addrspace(3) flat address: low 32 bits == workgroup-relative LDS offset
    const unsigned lds_base = (unsigned)(uintptr_t)wls;

    // ---- stage [0, cached) into LDS via the async DMA path (ASYNCcnt) ----
    {
        unsigned la = lds_base + (unsigned)((phase * D + dofs) * 4);
        unsigned long long ga = (unsigned long long)(uintptr_t)gp;
        int b = 0;
        for (; b < full; b += 4) {
            asm volatile("global_load_async_to_lds_b128 %0, %1, off"
                         :
                         : "v"(la), "v"(ga), "v"((void*)wls)
                         : "memory");
            la += 4u * 4u * D;              // 4 rows * 128B
            ga += 4ull * 4ull * D;
        }
        if (b + phase < cached) {           // partial last group (EXEC-masked)
            asm volatile("global_load_async_to_lds_b128 %0, %1, off"
                         :
                         : "v"(la), "v"(ga), "v"((void*)wls)
                         : "memory");
        }
    }
    asm volatile("s_wait_asynccnt 0" ::: "memory");

    // ---- pass 1: per-dim max over segment ----
    f32x4 m = make4(-3.402823466e38f);
    {
        int b = 0;
        for (; b < full; b += 4)
            m = vmax4(m, *(const f32x4*)(lp + (size_t)b * D));
        if (b + phase < cached)
            m = vmax4(m, *(const f32x4*)(lp + (size_t)b * D));
        for (b = CAP; b < len; b += 4)      // cold global tail (len > CAP only)
            if (b + phase < len)
                m = vmax4(m, *(const f32x4*)(gp + (size_t)b * D));
    }
    m = vmax4(m, shflxor4(m, 8));
    m = vmax4(m, shflxor4(m, 16));

    // ---- pass 2: per-dim sum of exp(x - max) ----
    f32x4 s = make4(0.0f);
    {
        int b = 0;
        for (; b < full; b += 4)
            s += vexp4(*(const f32x4*)(lp + (size_t)b * D) - m);
        if (b + phase < cached)
            s += vexp4(*(const f32x4*)(lp + (size_t)b * D) - m);
        for (b = CAP; b < len; b += 4)
            if (b + phase < len)
                s += vexp4(*(const f32x4*)(gp + (size_t)b * D) - m);
    }
    s += shflxor4(s, 8);
    s += shflxor4(s, 16);
    const f32x4 rinv = make4(1.0f) / s;     // s >= 1 per dim (max element contributes 1)

    // ---- pass 3: out = exp(x - max) / sum, streaming (non-temporal) b128 stores ----
    {
        float* op = out + ((size_t)(start + phase)) * D + dofs;
        int b = 0;
        for (; b < full; b += 4) {
            f32x4 o = vexp4(*(const f32x4*)(lp + (size_t)b * D) - m) * rinv;
            __builtin_nontemporal_store(o, (f32x4*)(op + (size_t)b * D));
        }
        if (b + phase < cached) {
            f32x4 o = vexp4(*(const f32x4*)(lp + (size_t)b * D) - m) * rinv;
            __builtin_nontemporal_store(o, (f32x4*)(op + (size_t)b * D));
        }
        for (b = CAP; b < len; b += 4) {
            if (b + phase < len) {
                f32x4 o = vexp4(*(const f32x4*)(gp + (size_t)b * D) - m) * rinv;
                __builtin_nontemporal_store(o, (f32x4*)(op + (size_t)b * D));
            }
        }
    }
}

// ---------------------------------------------------------------------------
extern "C" void kernel_launch(void* const* d_in, const int* in_sizes, int n_in,
                              void* d_out, int out_size, void* d_ws, size_t ws_size,
                              hipStream_t stream) {
    const float* edge_vec = (const float*)d_in[0];
    const int*   sel      = (const int*)d_in[1];
    float*       out      = (float*)d_out;
    int*         seg      = (int*)d_ws;     // N_NODES + 1 ints

    k_seg_offsets<<<(N_EDGES + 255) / 256, 256, 0, stream>>>(sel, seg);

    const int grid = (N_NODES + WAVES_PER_BLOCK - 1) / WAVES_PER_BLOCK;
    k_seg_softmax<<<grid, WAVES_PER_BLOCK * 32, 0, stream>>>(edge_vec, seg, out);

    (void)in_sizes; (void)n_in; (void)out_size; (void)ws_size;
}